// RGCN_721554506439
// MI455X (gfx1250) — compile-verified
//
#include <hip/hip_runtime.h>

#define NN 50000      // N_NODES
#define NE 600000     // N_EDGES
#define IND 128       // IN_DIM
#define HD 128        // H_DIM
#define OD 64         // OUT_DIM
#define NR 8          // NUM_RELS

typedef __attribute__((ext_vector_type(2))) float v2f;
typedef __attribute__((ext_vector_type(8))) float v8f;

// ---------------------------------------------------------------- utilities
__global__ void zero_f32(float* __restrict__ p, int n) {
  int i = blockIdx.x * blockDim.x + threadIdx.x;
  if (i < n) p[i] = 0.0f;
}

// per-(dst, rel) in-degree counts
__global__ void count_deg(const int* __restrict__ dst, const int* __restrict__ et,
                          float* __restrict__ deg, int n_edges) {
  int e = blockIdx.x * blockDim.x + threadIdx.x;
  if (e < n_edges) unsafeAtomicAdd(&deg[dst[e] * NR + et[e]], 1.0f);
}

// ------------------------------------------------------------- WMMA GEMM
// D[M x N] (+)= A[M x 128] * B[128 x N] (+bias) (relu), N = NT*16, K = 128.
// 256 threads = 8 waves; each wave computes one 16-row tile across all N-tiles.
// B is transposed into LDS ([n][k]) so B fragments are contiguous b64 loads.
template <int NT, bool ACCUM, bool RELU, bool BIAS>
__global__ __launch_bounds__(256) void gemm_wmma(const float* __restrict__ A,
                                                 const float* __restrict__ B,
                                                 float* __restrict__ D,
                                                 const float* __restrict__ bias,
                                                 int M) {
  constexpr int N = NT * 16;
  __shared__ float wt[N * 128];  // B transposed: wt[n*128 + k]

  const int tid = threadIdx.x;
  for (int i = tid; i < 128 * N; i += 256) {
    int k = i / N, n = i % N;          // B is row-major [K][N]
    wt[n * 128 + k] = B[i];
  }
  __syncthreads();

  const int lane = tid & 31;
  const int wave = tid >> 5;
  const int mtile = blockIdx.x * 8 + wave;
  if (mtile * 16 >= M) return;                 // wave-uniform: EXEC all-1s for WMMA
  const int m0 = mtile * 16;
  const int half = lane >> 4;                  // lane half selects K pair / row block
  const int mr = lane & 15;                    // A-fragment row
  const int col = lane & 15;                   // B/C fragment column within tile

  v8f zero = {};
  v8f acc[NT];
#pragma unroll
  for (int nt = 0; nt < NT; ++nt) acc[nt] = zero;

  // A fragment base: lane (0-15) reads K=k0,k0+1 of row mr; lane (16-31) K=k0+2,k0+3
  const float* arow = A + (size_t)(m0 + mr) * 128 + 2 * half;

#pragma unroll 4
  for (int kk = 0; kk < 32; ++kk) {
    const int k0 = kk * 4;
    v2f a = *(const v2f*)(arow + k0);
#pragma unroll
    for (int nt = 0; nt < NT; ++nt) {
      v2f b = *(const v2f*)(&wt[(nt * 16 + col) * 128 + k0 + 2 * half]);
      acc[nt] = __builtin_amdgcn_wmma_f32_16x16x4_f32(
          /*neg_a=*/false, a, /*neg_b=*/false, b,
          /*c_mod=*/(short)0, acc[nt], /*reuse_a=*/false, /*reuse_b=*/false);
    }
  }

  // C/D layout: VGPR j holds row j (lanes 0-15) / row j+8 (lanes 16-31), col = lane&15
#pragma unroll
  for (int nt = 0; nt < NT; ++nt) {
    const int cg = nt * 16 + col;
#pragma unroll
    for (int j = 0; j < 8; ++j) {
      const int row = m0 + j + half * 8;
      const size_t idx = (size_t)row * N + cg;
      float v = acc[nt][j];
      if (ACCUM) v += D[idx];
      if (BIAS) v += bias[cg];
      if (RELU) v = fmaxf(v, 0.0f);
      D[idx] = v;
    }
  }
}

// --------------------------------------------------- per-edge scatter-add
// one wave per edge; lanes stripe the feature row; per-(dst,rel) mean norm
template <int W>
__global__ __launch_bounds__(256) void scatter_edges(const float* __restrict__ xs,
                                                     const int* __restrict__ src,
                                                     const int* __restrict__ dst,
                                                     const int* __restrict__ et,
                                                     const float* __restrict__ deg,
                                                     float* __restrict__ out,
                                                     int rel, int n_edges) {
  const int lane = threadIdx.x & 31;
  const int e = blockIdx.x * 8 + (threadIdx.x >> 5);
  if (e >= n_edges) return;
  if (et[e] != rel) return;
  const int s = src[e];
  const int d = dst[e];
  const float norm = 1.0f / fmaxf(deg[d * NR + rel], 1.0f);
  const float* __restrict__ srow = xs + (size_t)s * W;
  float* __restrict__ drow = out + (size_t)d * W;
#pragma unroll
  for (int c = lane; c < W; c += 32) unsafeAtomicAdd(&drow[c], srow[c] * norm);
}

// ---------------------------------------------------------------- driver
extern "C" void kernel_launch(void* const* d_in, const int* in_sizes, int n_in,
                              void* d_out, int out_size, void* d_ws, size_t ws_size,
                              hipStream_t stream) {
  const float* x = (const float*)d_in[0];
  const int* eidx = (const int*)d_in[1];   // [2, NE]
  const int* et = (const int*)d_in[2];
  const float* W1 = (const float*)d_in[3];     // [8,128,128]
  const float* root1 = (const float*)d_in[4];  // [128,128]
  const float* b1 = (const float*)d_in[5];     // [128]
  const float* W2 = (const float*)d_in[6];     // [8,128,64]
  const float* root2 = (const float*)d_in[7];  // [128,64]
  const float* b2 = (const float*)d_in[8];     // [64]
  const int* src = eidx;
  const int* dst = eidx + NE;
  float* out = (float*)d_out;

  // workspace layout (floats): xs [NN*128] | h [NN*128] | deg [NN*8]  (~53 MB)
  float* xs = (float*)d_ws;
  float* h = xs + (size_t)NN * HD;
  float* deg = h + (size_t)NN * HD;

  const int ZB = 256;
  zero_f32<<<(NN * NR + ZB - 1) / ZB, ZB, 0, stream>>>(deg, NN * NR);
  zero_f32<<<(NN * HD + ZB - 1) / ZB, ZB, 0, stream>>>(h, NN * HD);
  count_deg<<<(NE + ZB - 1) / ZB, ZB, 0, stream>>>(dst, et, deg, NE);

  const int mtiles = NN / 16;                   // 3125 (exact)
  const int gblocks = (mtiles + 7) / 8;         // 8 waves/block
  const int eblocks = (NE + 7) / 8;             // 8 edges/block (1 wave each)

  // ---------------- layer 1: h = relu( sum_r mean_r(W1_r x) + x@root1 + b1 )
  for (int r = 0; r < NR; ++r) {
    gemm_wmma<8, false, false, false><<<gblocks, 256, 0, stream>>>(
        x, W1 + (size_t)r * IND * HD, xs, nullptr, NN);
    scatter_edges<HD><<<eblocks, 256, 0, stream>>>(xs, src, dst, et, deg, h, r, NE);
  }
  gemm_wmma<8, true, true, true><<<gblocks, 256, 0, stream>>>(x, root1, h, b1, NN);

  // ---------------- layer 2: out = sum_r mean_r(W2_r h) + h@root2 + b2
  zero_f32<<<(NN * OD + ZB - 1) / ZB, ZB, 0, stream>>>(out, NN * OD);
  for (int r = 0; r < NR; ++r) {
    gemm_wmma<4, false, false, false><<<gblocks, 256, 0, stream>>>(
        h, W2 + (size_t)r * HD * OD, xs, nullptr, NN);
    scatter_edges<OD><<<eblocks, 256, 0, stream>>>(xs, src, dst, et, deg, out, r, NE);
  }
  gemm_wmma<4, true, false, true><<<gblocks, 256, 0, stream>>>(h, root2, out, b2, NN);
}